// zdna_binder_8186207666385
// MI455X (gfx1250) — compile-verified
//
#include <hip/hip_runtime.h>
#include <math.h>

typedef float v2f __attribute__((ext_vector_type(2)));
typedef float v8f __attribute__((ext_vector_type(8)));

#define NB_GATHER 256
#define NT 256
#define NB_PASS 64
#define NWAVES (NB_PASS * NT / 32) /* 512 waves */
#define MAXIT 10

// ---------------------------------------------------------------------------
// Pass 0: gather CA coords (xyz[:,1,:]) + per-block f64 sum / sumsq partials.
// ---------------------------------------------------------------------------
__global__ void k_gather(const float* __restrict__ xyz, float* __restrict__ compact,
                         double* __restrict__ stats, int N)
{
    double sx = 0, sy = 0, sz = 0, sxx = 0, syy = 0, szz = 0;
    int tid = blockIdx.x * blockDim.x + threadIdx.x;
    int stride = gridDim.x * blockDim.x;
    for (int i = tid; i < N; i += stride) {
        const float* p = xyz + (size_t)i * 81 + 3;
        float x = p[0], y = p[1], z = p[2];
        if (compact) {
            float* c = compact + (size_t)i * 3;
            c[0] = x; c[1] = y; c[2] = z;
        }
        sx += x; sy += y; sz += z;
        sxx += (double)x * x; syy += (double)y * y; szz += (double)z * z;
    }
    __shared__ double red[NT];
    double acc[6] = {sx, sy, sz, sxx, syy, szz};
    for (int c = 0; c < 6; ++c) {
        red[threadIdx.x] = acc[c];
        __syncthreads();
        for (int off = NT / 2; off > 0; off >>= 1) {
            if ((int)threadIdx.x < off) red[threadIdx.x] += red[threadIdx.x + off];
            __syncthreads();
        }
        if (threadIdx.x == 0) stats[blockIdx.x * 6 + c] = red[0];
        __syncthreads();
    }
}

// ---------------------------------------------------------------------------
// Main pass: accumulate E[q][0..2] = sum_p e_qp * Pn_p and S[q] = sum_p e_qp
// via V_WMMA_F32_16X16X4_F32:  C(16x16) += A(16x4: e values) * B(4x16: [x y z 1]).
// Layouts (wave32):
//   A: VGPR0 = lanes0-15:A[m][0], lanes16-31:A[m][2]; VGPR1 = A[m][1]/A[m][3]
//   B: VGPR0 = lanes0-15:B[0][n], lanes16-31:B[2][n]; VGPR1 = B[1][n]/B[3][n]
//   C: VGPR v = lanes0-15: C[v][n], lanes16-31: C[v+8][n]
// The wave-stride loop is driven by a readfirstlane'd (scalar) wave index so
// the back-edge is a scalar branch: EXEC is structurally all-1s at the WMMA.
// ---------------------------------------------------------------------------
__global__ void k_pass(const float* __restrict__ Pbase, int pstride,
                       const float* __restrict__ params,
                       const float* __restrict__ Q,
                       float* __restrict__ wpart, int N)
{
    const float inv2s2 = 1.0f / (2.0f * 5.0f * 5.0f + 1e-6f);
    int tid  = blockIdx.x * blockDim.x + threadIdx.x;
    int lane = threadIdx.x & 31;
    int wave = __builtin_amdgcn_readfirstlane(tid >> 5);  // scalar, uniform
    int row  = lane & 15;   // A-row (q) for A-operand lanes; B/C column otherwise
    int hi   = lane >> 4;   // which K pair this lane holds

    float mx = params[0], my = params[1], mz = params[2], isd = params[3];

    bool qok = row < 12;
    float qx = 0.f, qy = 0.f, qz = 0.f;
    if (qok) { qx = Q[row * 3 + 0]; qy = Q[row * 3 + 1]; qz = Q[row * 3 + 2]; }
    float qq = qx * qx + qy * qy + qz * qz;

    v8f c = {0.f, 0.f, 0.f, 0.f, 0.f, 0.f, 0.f, 0.f};
    int nsteps = (N + 3) >> 2;
    for (int s = wave; s < nsteps; s += NWAVES) {
        int p0 = s * 4;
        int ka = p0 + (hi ? 2 : 0);
        int kb = ka + 1;
        bool va = ka < N, vb = kb < N;
        int ia = va ? ka : N - 1;
        int ib = vb ? kb : N - 1;
        const float* pa = Pbase + (size_t)ia * pstride;
        const float* pb = Pbase + (size_t)ib * pstride;
        float ax = (pa[0] - mx) * isd, ay = (pa[1] - my) * isd, az = (pa[2] - mz) * isd;
        float bx = (pb[0] - mx) * isd, by = (pb[1] - my) * isd, bz = (pb[2] - mz) * isd;

        float d2a = qq + (ax * ax + ay * ay + az * az) - 2.0f * (qx * ax + qy * ay + qz * az);
        float d2b = qq + (bx * bx + by * by + bz * bz) - 2.0f * (qx * bx + qy * by + qz * bz);
        d2a = fmaxf(d2a, 0.0f);
        d2b = fmaxf(d2b, 0.0f);
        float ea = (qok && va) ? expf(-d2a * inv2s2) : 0.0f;
        float eb = (qok && vb) ? expf(-d2b * inv2s2) : 0.0f;
        v2f a; a.x = ea; a.y = eb;

        float bxv = (row == 0) ? ax : (row == 1) ? ay : (row == 2) ? az : (row == 3) ? 1.0f : 0.0f;
        float byv = (row == 0) ? bx : (row == 1) ? by : (row == 2) ? bz : (row == 3) ? 1.0f : 0.0f;
        if (!va) bxv = 0.0f;
        if (!vb) byv = 0.0f;
        v2f b; b.x = bxv; b.y = byv;

        c = __builtin_amdgcn_wmma_f32_16x16x4_f32(false, a, false, b, (short)0, c, false, false);
    }

    // Extract rows 0..11, cols 0..3 of C into this wave's partial slot.
    float* wp = wpart + (size_t)wave * 48;
    if (lane < 4) {
#pragma unroll
        for (int v = 0; v < 8; ++v) wp[v * 4 + lane] = c[v];       // rows 0..7
    }
    if (lane >= 16 && lane < 20) {
        int col = lane - 16;
#pragma unroll
        for (int v = 0; v < 4; ++v) wp[(8 + v) * 4 + col] = c[v];  // rows 8..11
    }
}

// ---------------------------------------------------------------------------
// 3x3 symmetric Jacobi eigensolver (f64) + det
// ---------------------------------------------------------------------------
__device__ void jacobi3(double A[3][3], double V[3][3])
{
    for (int sweep = 0; sweep < 30; ++sweep) {
        for (int m = 0; m < 3; ++m) {
            int p = (m == 2) ? 1 : 0;
            int q = (m == 0) ? 1 : 2;
            double apq = A[p][q];
            if (fabs(apq) < 1e-300) continue;
            double theta = (A[q][q] - A[p][p]) / (2.0 * apq);
            double tt = (theta >= 0.0 ? 1.0 : -1.0) / (fabs(theta) + sqrt(theta * theta + 1.0));
            double cc = 1.0 / sqrt(tt * tt + 1.0), ss = tt * cc;
            for (int k = 0; k < 3; ++k) {
                double akp = A[k][p], akq = A[k][q];
                A[k][p] = cc * akp - ss * akq;
                A[k][q] = ss * akp + cc * akq;
            }
            for (int k = 0; k < 3; ++k) {
                double apk = A[p][k], aqk = A[q][k];
                A[p][k] = cc * apk - ss * aqk;
                A[q][k] = ss * apk + cc * aqk;
            }
            for (int k = 0; k < 3; ++k) {
                double vkp = V[k][p], vkq = V[k][q];
                V[k][p] = cc * vkp - ss * vkq;
                V[k][q] = ss * vkp + cc * vkq;
            }
        }
    }
}

__device__ double det3(const double M[3][3])
{
    return M[0][0] * (M[1][1] * M[2][2] - M[1][2] * M[2][1])
         - M[0][1] * (M[1][0] * M[2][2] - M[1][2] * M[2][0])
         + M[0][2] * (M[1][0] * M[2][1] - M[1][1] * M[2][0]);
}

// ---------------------------------------------------------------------------
// mode 0: finalize binder stats -> params; normalize zdna -> Qcur.
// mode 1..10: reduce wave partials, weighted Kabsch (f64), update Qcur;
//             mode==10 writes the final scalar.
// ---------------------------------------------------------------------------
__global__ void k_math(int mode,
                       const double* __restrict__ stats,
                       const float* __restrict__ zdna,
                       float* __restrict__ params,
                       float* __restrict__ qcur,
                       const float* __restrict__ wpart,
                       float* __restrict__ out, int N)
{
    __shared__ double sh[48];
    int tid = threadIdx.x;

    if (mode == 0) {
        if (tid < 6) {
            double s = 0;
            for (int b = 0; b < NB_GATHER; ++b) s += stats[b * 6 + tid];
            sh[tid] = s;
        }
        __syncthreads();
        if (tid == 0) {
            double n = (double)N;
            double m[3], var[3];
            for (int c = 0; c < 3; ++c) m[c] = sh[c] / n;
            for (int c = 0; c < 3; ++c) {
                double v = (sh[3 + c] - n * m[c] * m[c]) / (n - 1.0);
                var[c] = v > 0.0 ? v : 0.0;
            }
            double sd = (sqrt(var[0]) + sqrt(var[1]) + sqrt(var[2])) / 3.0;
            if (sd < 1e-6) sd = 1e-6;
            params[0] = (float)m[0]; params[1] = (float)m[1];
            params[2] = (float)m[2]; params[3] = (float)(1.0 / sd);

            double zm[3] = {0, 0, 0};
            for (int q = 0; q < 12; ++q)
                for (int c = 0; c < 3; ++c) zm[c] += (double)zdna[q * 3 + c];
            for (int c = 0; c < 3; ++c) zm[c] /= 12.0;
            double zv[3] = {0, 0, 0};
            for (int q = 0; q < 12; ++q)
                for (int c = 0; c < 3; ++c) {
                    double d = (double)zdna[q * 3 + c] - zm[c];
                    zv[c] += d * d;
                }
            double zsd = (sqrt(zv[0] / 11.0) + sqrt(zv[1] / 11.0) + sqrt(zv[2] / 11.0)) / 3.0;
            if (zsd < 1e-6) zsd = 1e-6;
            double zi = 1.0 / zsd;
            for (int q = 0; q < 12; ++q)
                for (int c = 0; c < 3; ++c)
                    qcur[q * 3 + c] = (float)(((double)zdna[q * 3 + c] - zm[c]) * zi);
        }
        return;
    }

    if (tid < 48) {
        double s = 0;
        for (int w = 0; w < NWAVES; ++w) s += (double)wpart[w * 48 + tid];
        sh[tid] = s;
    }
    __syncthreads();
    if (tid != 0) return;

    double WP[12][3], r[12], wsum = 0;
    for (int q = 0; q < 12; ++q) {
        double Sq = sh[q * 4 + 3];
        double den = Sq + 1e-6;
        r[q] = Sq / den;
        for (int c = 0; c < 3; ++c) WP[q][c] = sh[q * 4 + c] / den;
        wsum += r[q];
    }
    if (wsum < 1e-6) wsum = 1e-6;

    double Qd[12][3], cP[3] = {0, 0, 0}, cQ[3] = {0, 0, 0};
    for (int q = 0; q < 12; ++q)
        for (int c = 0; c < 3; ++c) {
            Qd[q][c] = (double)qcur[q * 3 + c];
            cP[c] += WP[q][c];
            cQ[c] += r[q] * Qd[q][c];
        }
    for (int c = 0; c < 3; ++c) { cP[c] /= wsum; cQ[c] /= wsum; }

    // H = P_cent^T (W^T Q_cent) = sum_q WP_q (x) (Q_q - cQ)   (cP term cancels)
    double H[3][3] = {{0, 0, 0}, {0, 0, 0}, {0, 0, 0}};
    for (int q = 0; q < 12; ++q)
        for (int i = 0; i < 3; ++i)
            for (int j = 0; j < 3; ++j)
                H[i][j] += WP[q][i] * (Qd[q][j] - cQ[j]);
    for (int i = 0; i < 3; ++i) H[i][i] += 1e-6;

    // R = (H^T H)^(-1/2) H^T  ==  Vh^T U^T from svd(H)
    double K[3][3];
    for (int i = 0; i < 3; ++i)
        for (int j = 0; j < 3; ++j) {
            double s = 0;
            for (int k = 0; k < 3; ++k) s += H[k][i] * H[k][j];
            K[i][j] = s;
        }
    double V[3][3] = {{1, 0, 0}, {0, 1, 0}, {0, 0, 1}};
    jacobi3(K, V);
    double rl[3];
    for (int i = 0; i < 3; ++i) {
        double l = K[i][i] > 0.0 ? K[i][i] : 0.0;
        rl[i] = 1.0 / sqrt(l + 1e-300);
    }
    double IS[3][3];
    for (int i = 0; i < 3; ++i)
        for (int j = 0; j < 3; ++j) {
            double s = 0;
            for (int k = 0; k < 3; ++k) s += V[i][k] * V[j][k] * rl[k];
            IS[i][j] = s;
        }
    double R[3][3];
    for (int i = 0; i < 3; ++i)
        for (int j = 0; j < 3; ++j) {
            double s = 0;
            for (int k = 0; k < 3; ++k) s += IS[i][k] * H[j][k];
            R[i][j] = s;
        }
    if (det3(R) < 0.0)
        for (int i = 0; i < 3; ++i)
            for (int j = 0; j < 3; ++j) R[i][j] = -R[i][j];

    double t[3];
    for (int j = 0; j < 3; ++j)
        t[j] = cQ[j] - (cP[0] * R[0][j] + cP[1] * R[1][j] + cP[2] * R[2][j]);

    double res2 = 0;
    float qn[36];
    for (int q = 0; q < 12; ++q)
        for (int j = 0; j < 3; ++j) {
            double v = Qd[q][0] * R[0][j] + Qd[q][1] * R[1][j] + Qd[q][2] * R[2][j] + t[j];
            qn[q * 3 + j] = (float)v;
            double dd = v - WP[q][j];
            res2 += dd * dd;
        }
    double rmsd = sqrt(res2 / 12.0 + 1e-6);
    for (int k = 0; k < 36; ++k) qcur[k] = qn[k];
    if (mode == MAXIT) out[0] = (float)(-10.0 * rmsd);
}

// ---------------------------------------------------------------------------
extern "C" void kernel_launch(void* const* d_in, const int* in_sizes, int n_in,
                              void* d_out, int out_size, void* d_ws, size_t ws_size,
                              hipStream_t stream)
{
    const float* xyz  = (const float*)d_in[0];
    const float* zdna = (const float*)d_in[1];
    int N = in_sizes[0] / 81;   // (N,27,3) flat

    char* ws = (char*)d_ws;
    size_t off_stats   = 0;
    size_t off_params  = off_stats + (size_t)NB_GATHER * 6 * sizeof(double);
    size_t off_q       = off_params + 64;
    size_t off_wpart   = off_q + 192;
    size_t off_compact = off_wpart + (size_t)NWAVES * 48 * sizeof(float);
    size_t need        = off_compact + (size_t)N * 3 * sizeof(float);
    bool use_compact   = (ws_size >= need);

    double* stats  = (double*)(ws + off_stats);
    float* params  = (float*)(ws + off_params);
    float* qcur    = (float*)(ws + off_q);
    float* wpart   = (float*)(ws + off_wpart);
    float* compact = use_compact ? (float*)(ws + off_compact) : nullptr;
    const float* Pbase = use_compact ? (const float*)compact : (xyz + 3);
    int pstride = use_compact ? 3 : 81;
    float* out = (float*)d_out;

    k_gather<<<NB_GATHER, NT, 0, stream>>>(xyz, compact, stats, N);
    k_math<<<1, 64, 0, stream>>>(0, stats, zdna, params, qcur, wpart, out, N);
    for (int it = 1; it <= MAXIT; ++it) {
        k_pass<<<NB_PASS, NT, 0, stream>>>(Pbase, pstride, params, qcur, wpart, N);
        k_math<<<1, 64, 0, stream>>>(it, stats, zdna, params, qcur, wpart, out, N);
    }
}